// SSIMLoss_24507083391563
// MI455X (gfx1250) — compile-verified
//
#include <hip/hip_runtime.h>
#include <stdint.h>

// ---------------------------------------------------------------------------
// SSIM loss, fused, CDNA5 (gfx1250).
//   * 11-tap separable Gaussian conv as banded GEMMs on V_WMMA_F32_16X16X4_F32
//     (fp32-exact matrix pipe), wave32 fragment layouts per ISA 7.12.2.
//   * Gaussian band fragments precomputed once per lane from a zero-padded
//     LDS table -> branch-free inner loops (pure ds_load + v_wmma).
//   * Interior tiles: halo tile DMA'd into LDS by the Tensor Data Mover
//     (tensor_load_to_lds + s_wait_tensorcnt); edge tiles: manual zero-padded
//     cooperative load.
//   * 5 channels {x, y, x^2, y^2, xy}, SSIM map, wave32 shuffle + LDS + atomic
//     reduction to scalar.
// ---------------------------------------------------------------------------

typedef float    v2f __attribute__((ext_vector_type(2)));
typedef float    v8f __attribute__((ext_vector_type(8)));
typedef uint32_t v4u __attribute__((ext_vector_type(4)));
typedef int      v4i __attribute__((ext_vector_type(4)));
typedef int      v8i __attribute__((ext_vector_type(8)));

#define TB 32            // output tile 32x32
#define CORE 42          // 32 + 10 halo
#define XS_ROWS 49       // 42 real rows + 7 zeroed (bands read rows up to 47, +1 spill)
#define HC_ROWS 48
#define HC_C 36          // 32 cols + stagger pad
#define NW 4
#define NTHREADS 128

// 11-tap Gaussian, sigma = 1.5, normalized (precomputed in double).
__device__ __constant__ float GW[11] = {
    0.00102840f, 0.00759876f, 0.03600077f, 0.10936071f, 0.21300553f,
    0.26601231f,
    0.21300553f, 0.10936071f, 0.03600077f, 0.00759876f, 0.00102840f};

__global__ void ssim_init_kernel(float* ws) { ws[0] = 0.0f; }

__global__ void ssim_final_kernel(const float* ws, float* out, float inv_n) {
    out[0] = 1.0f - ws[0] * inv_n;
}

#if __has_builtin(__builtin_amdgcn_tensor_load_to_lds)
#define HAVE_TDM 1
// 2D fp32 tile load, 42 rows x 42 cols, row stride = `stride` elements,
// packed densely into LDS at byte offset lds_off. Descriptor per ISA 8.3/8.4.
// 6-arg builtin form: (g0 x4, g1 x8, g2 x4, g3 x4, extra x8, cpol).
__device__ __forceinline__ void tdm_load_tile(uint64_t gaddr, uint32_t lds_off,
                                              uint32_t stride) {
    v4u g0;
    g0[0] = 1u;                                        // count=1, user D#
    g0[1] = lds_off;                                   // lds_addr
    g0[2] = (uint32_t)(gaddr & 0xFFFFFFFFu);           // global_addr[31:0]
    g0[3] = (uint32_t)((gaddr >> 32) & 0x01FFFFFFu)    // global_addr[56:32]
            | (2u << 30);                              // type = 2 (image)
    v8i g1;
    g1[0] = 0x20000;                                   // data_size=2 (4B)
    g1[1] = (int)0xFFFF0000u;                          // tensor_dim0 lo16 (huge)
    g1[2] = (int)0xFFFF7FFFu;                          // dim0 hi16 | dim1 lo16
    g1[3] = (int)((42u << 16) | 0x7FFFu);              // dim1 hi16 | tile_dim0=42
    g1[4] = 42;                                        // tile_dim1=42, tile_dim2=0
    g1[5] = (int)stride;                               // tensor_dim0_stride lo32
    g1[6] = 0;                                         // stride0 hi | stride1 lo
    g1[7] = 0;
    const v4i z4 = {0, 0, 0, 0};                       // 2D: groups 2/3 unused
    const v8i z8 = {0, 0, 0, 0, 0, 0, 0, 0};
    __builtin_amdgcn_tensor_load_to_lds(g0, g1, z4, z4, z8, 0);
}
#else
#define HAVE_TDM 0
#endif

__global__ __launch_bounds__(NTHREADS)
void ssim_tile_kernel(const float* __restrict__ x, const float* __restrict__ y,
                      float* __restrict__ ws_sum, int H, int W)
{
    __shared__ float xs[XS_ROWS][CORE];        // dense stride 42 (TDM layout)
    __shared__ float ys[XS_ROWS][CORE];
    __shared__ float hc[5][HC_ROWS][HC_C];     // H-convolved channels
    __shared__ float gwt[44];                  // zero-padded gaussian: gwt[k+16]
    __shared__ float red[NW];

    const int tid   = threadIdx.x;
    const int wave  = tid >> 5;
    const int lane  = tid & 31;
    const int rbase = blockIdx.y * TB;
    const int cbase = blockIdx.x * TB;
    const long imgOff = (long)blockIdx.z * H * W;

    // zero-padded gaussian table: gwt[i] = GW[i-16] for i in [16,27), else 0
    if (tid < 44) gwt[tid] = (tid >= 16 && tid < 27) ? GW[tid - 16] : 0.0f;

    // ---- Stage 1: halo tile (42x42) into LDS ----
    bool tdm_path = false;
#if HAVE_TDM
    tdm_path = (blockIdx.x >= 1) && (blockIdx.x + 2 <= gridDim.x) &&
               (blockIdx.y >= 1) && (blockIdx.y + 2 <= gridDim.y);
    if (tdm_path) {
        if (wave == 0) {   // one TDM issue per block; EXEC ignored by TDM
            const long org = imgOff + (long)(rbase - 5) * W + (cbase - 5);
            tdm_load_tile((uint64_t)(uintptr_t)(x + org),
                          (uint32_t)(uintptr_t)&xs[0][0], (uint32_t)W);
            tdm_load_tile((uint64_t)(uintptr_t)(y + org),
                          (uint32_t)(uintptr_t)&ys[0][0], (uint32_t)W);
            __builtin_amdgcn_s_wait_tensorcnt(0);
        }
    }
#endif
    if (!tdm_path) {       // edge tiles: manual zero-padded load
        for (int idx = tid; idx < CORE * CORE; idx += NTHREADS) {
            const int lr = idx / CORE, lc = idx % CORE;
            const int gr = rbase + lr - 5, gc = cbase + lc - 5;
            float xv = 0.0f, yv = 0.0f;
            if (gr >= 0 && gr < H && gc >= 0 && gc < W) {
                const long o = imgOff + (long)gr * W + gc;
                xv = x[o];
                yv = y[o];
            }
            xs[lr][lc] = xv;
            ys[lr][lc] = yv;
        }
    }
    // zero rows 42..48 (band tail + spill row) -- keeps 0-weight lanes NaN-free
    for (int idx = tid; idx < (XS_ROWS - CORE) * CORE; idx += NTHREADS) {
        const int lr = CORE + idx / CORE, lc = idx % CORE;
        xs[lr][lc] = 0.0f;
        ys[lr][lc] = 0.0f;
    }
    __syncthreads();

    // Per-lane WMMA fragment coordinates (ISA 7.12.2, wave32):
    //  A 16x4 fp32: lanes 0-15 hold K=0(v0),1(v1); lanes 16-31 hold K=2,3.
    const int m   = lane & 15;            // M row (A/D) == N col (B/D)
    const int kb  = (lane >> 4) << 1;     // K base for this half-wave: 0 or 2
    const int hi8 = (lane >> 4) << 3;     // D rows: VGPR i -> M = i (+8 hi half)

    // Band-matrix fragments: identical for H-pass B and V-pass A:
    //   frag[s] = { g[4s+kb - m], g[4s+kb+1 - m] }   (zero outside [0,10])
    v2f wfrag[7];
#pragma unroll
    for (int s = 0; s < 7; ++s) {
        const int kk = 4 * s + kb;
        wfrag[s].x = gwt[kk - m + 16];
        wfrag[s].y = gwt[kk + 1 - m + 16];
    }

    // ---- Stage 2: horizontal conv, 6 geometry tasks x 5 channel accs ----
    for (int g = wave; g < 6; g += NW) {
        const int r0 = (g >> 1) << 4;     // 0,16,32
        const int c0 = (g & 1) << 4;      // 0,16
        v8f a0 = {0.f,0.f,0.f,0.f,0.f,0.f,0.f,0.f};
        v8f a1 = a0, a2 = a0, a3 = a0, a4 = a0;
        const float* xr = &xs[r0 + m][c0];
        const float* yr = &ys[r0 + m][c0];
        for (int s = 0; s < 7; ++s) {     // K = 26 (7 slices of 4, zero-padded)
            const int kk = 4 * s + kb;
            const float x0 = xr[kk], x1 = xr[kk + 1];
            const float y0 = yr[kk], y1 = yr[kk + 1];
            const v2f b = wfrag[s];
            v2f a;
            a.x = x0;      a.y = x1;
            a0 = __builtin_amdgcn_wmma_f32_16x16x4_f32(false, a, false, b, (short)0, a0, false, false);
            a.x = y0;      a.y = y1;
            a1 = __builtin_amdgcn_wmma_f32_16x16x4_f32(false, a, false, b, (short)0, a1, false, false);
            a.x = x0 * x0; a.y = x1 * x1;
            a2 = __builtin_amdgcn_wmma_f32_16x16x4_f32(false, a, false, b, (short)0, a2, false, false);
            a.x = y0 * y0; a.y = y1 * y1;
            a3 = __builtin_amdgcn_wmma_f32_16x16x4_f32(false, a, false, b, (short)0, a3, false, false);
            a.x = x0 * y0; a.y = x1 * y1;
            a4 = __builtin_amdgcn_wmma_f32_16x16x4_f32(false, a, false, b, (short)0, a4, false, false);
        }
#pragma unroll
        for (int i = 0; i < 8; ++i) {
            const int r = r0 + i + hi8;
            hc[0][r][c0 + m] = a0[i];
            hc[1][r][c0 + m] = a1[i];
            hc[2][r][c0 + m] = a2[i];
            hc[3][r][c0 + m] = a3[i];
            hc[4][r][c0 + m] = a4[i];
        }
    }
    __syncthreads();

    // ---- Stage 3: vertical conv, one 16x16 output subtile per wave ----
    const int m0 = (wave & 1) << 4;
    const int n0 = (wave >> 1) << 4;
    v8f accs[5];
#pragma unroll
    for (int c = 0; c < 5; ++c)
        accs[c] = (v8f){0.f,0.f,0.f,0.f,0.f,0.f,0.f,0.f};

    for (int s = 0; s < 7; ++s) {
        const int kk = 4 * s + kb;
        const v2f a = wfrag[s];           // band: A[m][kk] = g[kk - m]
#pragma unroll
        for (int c = 0; c < 5; ++c) {
            v2f b;
            b.x = hc[c][m0 + kk][n0 + m];
            b.y = hc[c][m0 + kk + 1][n0 + m];
            accs[c] = __builtin_amdgcn_wmma_f32_16x16x4_f32(false, a, false, b, (short)0, accs[c], false, false);
        }
    }

    // ---- SSIM map terms + reduction ----
    const float C1f = 1.0e-4f;   // 0.01^2
    const float C2f = 9.0e-4f;   // 0.03^2
    float lsum = 0.0f;
#pragma unroll
    for (int i = 0; i < 8; ++i) {
        const float mx  = accs[0][i], my  = accs[1][i];
        const float exx = accs[2][i], eyy = accs[3][i], exy = accs[4][i];
        const float mx2 = mx * mx, my2 = my * my, mxy = mx * my;
        const float sx  = exx - mx2;
        const float sy  = eyy - my2;
        const float sxy = exy - mxy;
        const float num = (2.0f * mxy + C1f) * (2.0f * sxy + C2f);
        const float den = (mx2 + my2 + C1f) * (sx + sy + C2f);
        lsum += num / den;
    }
    for (int off = 16; off > 0; off >>= 1)
        lsum += __shfl_down(lsum, off, 32);
    if (lane == 0) red[wave] = lsum;
    __syncthreads();
    if (tid == 0) {
        float bsum = 0.0f;
#pragma unroll
        for (int w = 0; w < NW; ++w) bsum += red[w];
        atomicAdd(ws_sum, bsum);
    }
}

extern "C" void kernel_launch(void* const* d_in, const int* in_sizes, int n_in,
                              void* d_out, int out_size, void* d_ws, size_t ws_size,
                              hipStream_t stream) {
    const float* x = (const float*)d_in[0];
    const float* y = (const float*)d_in[1];
    float* out = (float*)d_out;
    float* ws  = (float*)d_ws;

    const int H = 512, W = 512;
    const int B = in_sizes[0] / (H * W);   // 32

    ssim_init_kernel<<<1, 1, 0, stream>>>(ws);

    dim3 grid(W / TB, H / TB, B);          // 16 x 16 x 32 = 8192 blocks
    ssim_tile_kernel<<<grid, NTHREADS, 0, stream>>>(x, y, ws, H, W);

    const float inv_n = 1.0f / ((float)B * (float)H * (float)W);
    ssim_final_kernel<<<1, 1, 0, stream>>>(ws, out, inv_n);
}